// EmbeddingLayer_33036888441229
// MI455X (gfx1250) — compile-verified
//
#include <hip/hip_runtime.h>

// Embedding gather: out[b,s,:] = W[:, x[b,s]] + b   (W is [EMBED, VOCAB])
// Strategy: (1) fuse bias into a tiled transpose W -> Wt[v][e] in d_ws,
//           (2) wave32-per-token row copy (32 lanes x 16B = 512B row).
// MI455X-specific: output stream uses non-temporal stores so the 25.7MB Wt
// table stays resident in the 192MB L2 while 134MB of output streams past it.

typedef float v4f __attribute__((ext_vector_type(4))); // native vec for NT ops

#define VOCAB 50257
#define EMBED 128
#define NTOK  (64 * 4096) /* 262144 tokens */

#define TILE_V 32

// ---------------------------------------------------------------------------
// Pass 1: Wt[v][e] = W[e][v] + b[e]
// Reads W coalesced along vocab (128B per row-segment) with NT hint (read
// once), writes Wt as contiguous 512B rows with regular temporal hint (it is
// re-read by pass 2 and must live in L2). LDS tile [EMBED][TILE_V+1]
// (pad 1 -> fewer bank conflicts on the column reads on the 64-bank LDS).
// ---------------------------------------------------------------------------
__global__ __launch_bounds__(256) void wt_transpose_bias(
    const float* __restrict__ W, const float* __restrict__ bias,
    float* __restrict__ Wt) {
  __shared__ float tile[EMBED][TILE_V + 1]; // 16.9 KB of the 320 KB LDS

  const int tid  = threadIdx.x;  // 0..255
  const int lane = tid & 31;     // wave32 lane
  const int wv   = tid >> 5;     // wave id 0..7
  const int v0   = blockIdx.x * TILE_V;

  // Per-lane bias chunk: lane covers e = 4*lane .. 4*lane+3 (coalesced 512B).
  const v4f bvec = *reinterpret_cast<const v4f*>(bias + lane * 4);

  // Load: lanes sweep vocab (coalesced); waves sweep embed rows.
  // Pointer-walk (step = 8*VOCAB floats) instead of per-step multiplies.
  const int v = v0 + lane;
  if (v < VOCAB) {
    const float* p = W + (size_t)wv * VOCAB + v;
#pragma unroll
    for (int e8 = 0; e8 < EMBED / 8; ++e8) {
      tile[wv + e8 * 8][lane] = __builtin_nontemporal_load(p);
      p += (size_t)8 * VOCAB;
    }
  }
  __syncthreads();

  // Store: each wave writes 4 vocab rows; per row each lane stores 16B
  // -> one contiguous 512B global write per wave. Regular temporal hint:
  // these lines are the gather pass's working set.
  for (int rr = 0; rr < 4; ++rr) {
    const int row = wv * 4 + rr; // 0..31
    const int vv  = v0 + row;
    if (vv < VOCAB) {
      const int e = lane * 4;
      v4f o;
      o.x = tile[e + 0][row] + bvec.x;
      o.y = tile[e + 1][row] + bvec.y;
      o.z = tile[e + 2][row] + bvec.z;
      o.w = tile[e + 3][row] + bvec.w;
      *reinterpret_cast<v4f*>(Wt + (size_t)vv * EMBED + e) = o;
    }
  }
}

// ---------------------------------------------------------------------------
// Pass 2: out[t][:] = Wt[idx[t]][:]
// One wave32 per token (32 lanes x 16B = exactly one 512B row).
// 4 tokens per wave for ILP: 4 independent b128 load chains in flight.
// Indices fetched as one aligned int4 (single global_load_b128 per wave).
// Output written with NT stores so it does not evict Wt from L2.
// ---------------------------------------------------------------------------
#define TOK_PER_WAVE    4
#define WAVES_PER_BLOCK 8
#define TOK_PER_BLOCK   (TOK_PER_WAVE * WAVES_PER_BLOCK) // 32

__global__ __launch_bounds__(256) void embed_gather(
    const int* __restrict__ idx, const float* __restrict__ Wt,
    float* __restrict__ out) {
  const int tid  = threadIdx.x;
  const int lane = tid & 31;
  const int wv   = tid >> 5;
  const size_t t0 =
      (size_t)blockIdx.x * TOK_PER_BLOCK + (size_t)wv * TOK_PER_WAVE;

  // gfx1250 speculative prefetch of the index stream well ahead
  // (lowered to global_prefetch_b8; OOB speculative prefetch is dropped).
  __builtin_prefetch(idx + t0 + (size_t)TOK_PER_BLOCK * 64, 0, 0);

  // t0 is a multiple of 4 -> 16B-aligned int4 load of 4 consecutive indices.
  const int4 vi = *reinterpret_cast<const int4*>(idx + t0);
  const int v[TOK_PER_WAVE] = {vi.x, vi.y, vi.z, vi.w};

  v4f r[TOK_PER_WAVE];
#pragma unroll
  for (int k = 0; k < TOK_PER_WAVE; ++k)
    r[k] = reinterpret_cast<const v4f*>(Wt + (size_t)v[k] * EMBED)[lane];

#pragma unroll
  for (int k = 0; k < TOK_PER_WAVE; ++k)
    __builtin_nontemporal_store(
        r[k], reinterpret_cast<v4f*>(out + (t0 + k) * (size_t)EMBED) + lane);
}

// ---------------------------------------------------------------------------
// Fallback (only if d_ws can't hold Wt): direct strided gather.
// ---------------------------------------------------------------------------
__global__ __launch_bounds__(256) void embed_gather_direct(
    const int* __restrict__ idx, const float* __restrict__ W,
    const float* __restrict__ bias, float* __restrict__ out) {
  const int tid  = threadIdx.x;
  const int lane = tid & 31;
  const int wv   = tid >> 5;
  const size_t t = (size_t)blockIdx.x * WAVES_PER_BLOCK + wv;

  const int v = idx[t];
  const int e = lane * 4;
  const v4f bvec = *reinterpret_cast<const v4f*>(bias + e);
  v4f o;
  o.x = W[(size_t)(e + 0) * VOCAB + v] + bvec.x;
  o.y = W[(size_t)(e + 1) * VOCAB + v] + bvec.y;
  o.z = W[(size_t)(e + 2) * VOCAB + v] + bvec.z;
  o.w = W[(size_t)(e + 3) * VOCAB + v] + bvec.w;
  __builtin_nontemporal_store(
      o, reinterpret_cast<v4f*>(out + t * (size_t)EMBED + e));
}

extern "C" void kernel_launch(void* const* d_in, const int* in_sizes, int n_in,
                              void* d_out, int out_size, void* d_ws,
                              size_t ws_size, hipStream_t stream) {
  const int*   x = (const int*)d_in[0];   // [64, 4096] int32
  const float* W = (const float*)d_in[1]; // [128, 50257] f32
  const float* b = (const float*)d_in[2]; // [128] f32
  float* out = (float*)d_out;             // [64, 4096, 128] f32

  const size_t wt_bytes = (size_t)VOCAB * EMBED * sizeof(float); // 25.7 MB

  if (ws_size >= wt_bytes) {
    float* Wt = (float*)d_ws;
    const int tblocks = (VOCAB + TILE_V - 1) / TILE_V; // 1571
    wt_transpose_bias<<<tblocks, 256, 0, stream>>>(W, b, Wt);
    embed_gather<<<NTOK / TOK_PER_BLOCK, 256, 0, stream>>>(x, Wt, out);
  } else {
    embed_gather_direct<<<NTOK / WAVES_PER_BLOCK, 256, 0, stream>>>(x, W, b,
                                                                    out);
  }
}